// APPNP_15195594293933
// MI455X (gfx1250) — compile-verified
//
#include <hip/hip_runtime.h>
#include <hip/hip_bf16.h>

#define N_FEAT   512
#define N_HID    64
#define N_CLS    40
#define N_LAYERS 10
#define ALPHA    0.8f

typedef __attribute__((ext_vector_type(2))) float v2f;
typedef __attribute__((ext_vector_type(8))) float v8f;

// -------------------------------------------------------------------------
// h0 = x @ W_in + b_in        (one wave per 16x16 tile, V_WMMA_F32_16X16X4_F32)
// block = (32,4): 4 waves cover 16 rows x 64 cols. grid.x = nNodes/16.
// -------------------------------------------------------------------------
__global__ __launch_bounds__(128) void gemm_in_kernel(
    const float* __restrict__ x, const float* __restrict__ W,
    const float* __restrict__ bias, float* __restrict__ h)
{
    const int lane   = threadIdx.x;          // 0..31
    const int lane16 = lane & 15;
    const int half   = lane >> 4;            // 0: K=0,1   1: K=2,3
    const int row0   = blockIdx.x * 16;
    const int col0   = threadIdx.y * 16;

    const float* xrow = x + (size_t)(row0 + lane16) * N_FEAT + 2 * half;
    const float* wcol = W + col0 + lane16;

    v8f acc = {0.f, 0.f, 0.f, 0.f, 0.f, 0.f, 0.f, 0.f};
    for (int k = 0; k < N_FEAT; k += 4) {
        v2f a;
        a.x = xrow[k];                       // contiguous float2 per lane
        a.y = xrow[k + 1];
        const int ka = k + 2 * half;
        v2f b;
        b.x = wcol[(size_t)ka * N_HID];
        b.y = wcol[(size_t)(ka + 1) * N_HID];
        acc = __builtin_amdgcn_wmma_f32_16x16x4_f32(
            false, a, false, b, (short)0, acc, false, false);
    }

    const float bv = bias[col0 + lane16];
#pragma unroll
    for (int i = 0; i < 8; ++i) {
        const int m = i + 8 * half;          // C/D layout: VGPR i -> M=i / M=i+8
        h[(size_t)(row0 + m) * N_HID + col0 + lane16] = acc[i] + bv;
    }
}

// -------------------------------------------------------------------------
// out = h @ W_out + b_out     (N_CLS=40 -> 3 col tiles, mask cols >= 40)
// block = (32,3). grid.x = nNodes/16.
// -------------------------------------------------------------------------
__global__ __launch_bounds__(96) void gemm_out_kernel(
    const float* __restrict__ h, const float* __restrict__ W,
    const float* __restrict__ bias, float* __restrict__ out)
{
    const int lane   = threadIdx.x;
    const int lane16 = lane & 15;
    const int half   = lane >> 4;
    const int row0   = blockIdx.x * 16;
    const int col0   = threadIdx.y * 16;
    const int col    = col0 + lane16;
    const bool colOK = (col < N_CLS);

    const float* hrow = h + (size_t)(row0 + lane16) * N_HID + 2 * half;

    v8f acc = {0.f, 0.f, 0.f, 0.f, 0.f, 0.f, 0.f, 0.f};
    for (int k = 0; k < N_HID; k += 4) {
        const int ka = k + 2 * half;
        v2f a;
        a.x = hrow[k];
        a.y = hrow[k + 1];
        v2f b;                                // predicated loads (cndmask, EXEC stays full)
        b.x = colOK ? W[(size_t)ka * N_CLS + col] : 0.f;
        b.y = colOK ? W[(size_t)(ka + 1) * N_CLS + col] : 0.f;
        acc = __builtin_amdgcn_wmma_f32_16x16x4_f32(
            false, a, false, b, (short)0, acc, false, false);
    }

    const float bv = colOK ? bias[col] : 0.f;
#pragma unroll
    for (int i = 0; i < 8; ++i) {
        const int m = i + 8 * half;
        if (colOK)
            out[(size_t)(row0 + m) * N_CLS + col] = acc[i] + bv;
    }
}

// -------------------------------------------------------------------------
// h_next = (1 - alpha) * h0   (float4 vectorized; fully re-inits the buffer)
// -------------------------------------------------------------------------
__global__ __launch_bounds__(256) void scale_init_kernel(
    const float4* __restrict__ h0, float4* __restrict__ hout, int n4)
{
    const int i = blockIdx.x * blockDim.x + threadIdx.x;
    if (i < n4) {
        float4 v = h0[i];
        const float s = 1.0f - ALPHA;
        v.x *= s; v.y *= s; v.z *= s; v.w *= s;
        hout[i] = v;
    }
}

// -------------------------------------------------------------------------
// h_next[dst] += alpha * w[e] * h[src]
// 16 threads per edge, each thread owns 4 consecutive features:
//   - gather is one global_load_b128 per thread (512B/wave: 256B contiguous per edge)
//   - 4 no-return f32 atomic adds (relaxed/agent -> global_atomic_add_f32)
// Cuts per-edge vmem instruction count ~3x vs 1 feature/thread; the atomic
// count (64/edge) is the irreducible scatter cost and stays L2-resident.
// -------------------------------------------------------------------------
__global__ __launch_bounds__(256) void edge_scatter_kernel(
    const long long* __restrict__ ei, const float* __restrict__ ew,
    const float* __restrict__ hin, float* __restrict__ hout, int nE)
{
    const int t  = blockIdx.x * blockDim.x + threadIdx.x;
    const int e  = t >> 4;                   // 16 threads per edge
    const int f0 = (t & 15) * 4;             // 4 features per thread
    if (e >= nE) return;

    const int src = (int)ei[e];
    const int dst = (int)ei[(size_t)nE + e];
    const float w = ew[e] * ALPHA;

    const float4 hv = *(const float4*)(hin + (size_t)src * N_HID + f0);
    float* dp = hout + (size_t)dst * N_HID + f0;

    __hip_atomic_fetch_add(dp + 0, hv.x * w, __ATOMIC_RELAXED, __HIP_MEMORY_SCOPE_AGENT);
    __hip_atomic_fetch_add(dp + 1, hv.y * w, __ATOMIC_RELAXED, __HIP_MEMORY_SCOPE_AGENT);
    __hip_atomic_fetch_add(dp + 2, hv.z * w, __ATOMIC_RELAXED, __HIP_MEMORY_SCOPE_AGENT);
    __hip_atomic_fetch_add(dp + 3, hv.w * w, __ATOMIC_RELAXED, __HIP_MEMORY_SCOPE_AGENT);
}

// -------------------------------------------------------------------------
extern "C" void kernel_launch(void* const* d_in, const int* in_sizes, int n_in,
                              void* d_out, int out_size, void* d_ws, size_t ws_size,
                              hipStream_t stream)
{
    const float*     x     = (const float*)d_in[0];
    const long long* ei    = (const long long*)d_in[1];   // int64 edge_index [2][E]
    const float*     ew    = (const float*)d_in[2];
    const float*     W_in  = (const float*)d_in[3];
    const float*     b_in  = (const float*)d_in[4];
    const float*     W_out = (const float*)d_in[5];
    const float*     b_out = (const float*)d_in[6];
    float*           out   = (float*)d_out;

    const int nNodes = in_sizes[0] / N_FEAT;   // 50000 (multiple of 16)
    const int nE     = in_sizes[1] / 2;        // 800000

    // workspace: h0 | hA | hB  (each 12.8 MB, 256B aligned; all L2-resident)
    const size_t hBytes = ((size_t)nNodes * N_HID * sizeof(float) + 255) & ~(size_t)255;
    char*  ws = (char*)d_ws;
    float* h0 = (float*)(ws);
    float* hA = (float*)(ws + hBytes);
    float* hB = (float*)(ws + 2 * hBytes);

    // 1) h0 = x @ W_in + b_in
    {
        dim3 blk(32, 4);
        gemm_in_kernel<<<nNodes / 16, blk, 0, stream>>>(x, W_in, b_in, h0);
    }

    // 2) 10 propagation layers, ping-pong hA/hB
    const int n4         = nNodes * N_HID / 4;
    const int initBlocks = (n4 + 255) / 256;
    const int edgeBlocks = (nE * 16 + 255) / 256;   // 12.8M threads / layer

    const float* cur = h0;
    float*       nxt = hA;
    for (int l = 0; l < N_LAYERS; ++l) {
        scale_init_kernel<<<initBlocks, 256, 0, stream>>>(
            (const float4*)h0, (float4*)nxt, n4);
        edge_scatter_kernel<<<edgeBlocks, 256, 0, stream>>>(ei, ew, cur, nxt, nE);
        cur = nxt;
        nxt = (nxt == hA) ? hB : hA;
    }

    // 3) out = h @ W_out + b_out
    {
        dim3 blk(32, 3);
        gemm_out_kernel<<<nNodes / 16, blk, 0, stream>>>(cur, W_out, b_out, out);
    }
}